// Pooler_5222680232385
// MI455X (gfx1250) — compile-verified
//
#include <hip/hip_runtime.h>
#include <hip/hip_bf16.h>

typedef __attribute__((ext_vector_type(2))) float v2f;
typedef __attribute__((ext_vector_type(8))) float v8f;

#define NSAMP 196   // 14x14 sub-sample grid
#define CTILE 64    // channels per LDS tile
#define SPITCH 65   // pitch (floats) to avoid LDS bank conflicts

__global__ __launch_bounds__(256)
void roi_pool_fpn_wmma(const float* __restrict__ f0, const float* __restrict__ f1,
                       const float* __restrict__ f2, const float* __restrict__ f3,
                       const float* __restrict__ boxes, const int* __restrict__ bidx,
                       float* __restrict__ out, int nBoxes)
{
    __shared__ int   goff[NSAMP][4];     // flattened corner offsets (y*W+x)
    __shared__ float gwt [NSAMP][4];     // bilinear weights (masked by validity)
    __shared__ float S[NSAMP * SPITCH];  // sampled values, one 64-channel tile

    const int box = blockIdx.x;
    if (box >= nBoxes) return;           // uniform per block
    const int tid = threadIdx.x;

    const float bx1 = boxes[box*4+0], by1 = boxes[box*4+1];
    const float bx2 = boxes[box*4+2], by2 = boxes[box*4+3];

    // ---- level assignment ----
    const float wb = bx2 - bx1 + 1.0f, hb = by2 - by1 + 1.0f;
    const float sb = sqrtf(wb * hb);
    float lv = floorf(4.0f + log2f(1e-6f + sb * (1.0f / 224.0f)));
    lv = fminf(fmaxf(lv, 2.0f), 5.0f);
    const int lvl = (int)lv - 2;

    const float* feat; int H, W; float scale;
    switch (lvl) {
        case 0:  feat = f0; H = 200; W = 304; scale = 0.25f;    break;
        case 1:  feat = f1; H = 100; W = 152; scale = 0.125f;   break;
        case 2:  feat = f2; H = 50;  W = 76;  scale = 0.0625f;  break;
        default: feat = f3; H = 25;  W = 38;  scale = 0.03125f; break;
    }
    const int b = bidx[box];
    const size_t plane = (size_t)H * W;
    const float* fbase = feat + (size_t)b * 256 * plane;

    const float x1s = bx1 * scale, y1s = by1 * scale;
    const float roiW = fmaxf(bx2 * scale - x1s, 1.0f);
    const float roiH = fmaxf(by2 * scale - y1s, 1.0f);
    const float binW = roiW * (1.0f / 7.0f), binH = roiH * (1.0f / 7.0f);

    // ---- per-sample bilinear geometry (196 samples): offsets + weights ----
    if (tid < NSAMP) {
        const int j = tid / 14, i = tid % 14;             // j = y-sample, i = x-sample
        const float gy = (float)(j >> 1) + ((float)(j & 1) + 0.5f) * 0.5f;
        const float gx = (float)(i >> 1) + ((float)(i & 1) + 0.5f) * 0.5f;
        const float y = y1s + gy * binH;
        const float x = x1s + gx * binW;
        const bool valid = (y >= -1.0f) && (y <= (float)H) && (x >= -1.0f) && (x <= (float)W);
        const float yc = fminf(fmaxf(y, 0.0f), (float)(H - 1));
        const float xc = fminf(fmaxf(x, 0.0f), (float)(W - 1));
        const int iy0 = (int)floorf(yc), ix0 = (int)floorf(xc);
        const int iy1 = min(iy0 + 1, H - 1), ix1 = min(ix0 + 1, W - 1);
        const float ly = yc - (float)iy0, lx = xc - (float)ix0;
        const float hy = 1.0f - ly, hx = 1.0f - lx;
        const float m = valid ? 1.0f : 0.0f;
        goff[tid][0] = iy0 * W + ix0;  gwt[tid][0] = m * hy * hx;
        goff[tid][1] = iy0 * W + ix1;  gwt[tid][1] = m * hy * lx;
        goff[tid][2] = iy1 * W + ix0;  gwt[tid][2] = m * ly * hx;
        goff[tid][3] = iy1 * W + ix1;  gwt[tid][3] = m * ly * lx;
    }
    __syncthreads();

    const int wave = tid >> 5, lane = tid & 31;
    const int half = lane >> 4, l15 = lane & 15;

    for (int ct = 0; ct < 4; ++ct) {
        const int ctglob = ct * CTILE;

        // ---- gather phase: fill S[196 x 64]; lane-consecutive samples -> local loads.
        // 49*256 = 12544 items exactly; incremental s = w%196, cc = w/196.
        {
            int s = tid, cc = 0;
            if (s >= NSAMP) { s -= NSAMP; cc = 1; }
            #pragma unroll 1
            for (int it = 0; it < 49; ++it) {
                const float* fp = fbase + (size_t)(ctglob + cc) * plane;
                const float v = gwt[s][0] * fp[goff[s][0]]
                              + gwt[s][1] * fp[goff[s][1]]
                              + gwt[s][2] * fp[goff[s][2]]
                              + gwt[s][3] * fp[goff[s][3]];
                S[s * SPITCH + cc] = v;
                s += 60; cc += 1;
                if (s >= NSAMP) { s -= NSAMP; cc += 2 - 1; }
            }
        }
        __syncthreads();

        // ---- pooling phase: D(16 ch x 16 pix) = A(16x4) x B(4x16), K = 4 subsamples.
        // A = sampled values (from LDS); B = 0.25 one-hot selecting output pixel column.
        // 16 tasks = 4 channel-subtiles x 4 pixel-tiles; 8 waves x 2 tasks, constant trips.
        #pragma unroll
        for (int tt = 0; tt < 2; ++tt) {
            const int task    = wave + tt * 8;        // 0..15
            const int cbase   = (task >> 2) * 16;     // channel subtile within 64
            const int pixbase = (task & 3) * 16;      // pixel tile base (0,16,32,48)

            // lane-constant LDS offset: A rows = channels (M = l15), K pair by half:
            // s(q,k) = base(q) + {0,1,14,15}[k];  lanes 0-15: K0/K1, 16-31: K2/K3
            const int lbase = (half ? 14 * SPITCH : 0) + cbase + l15;

            v8f acc = {};
            #pragma unroll
            for (int q = 0; q < 16; ++q) {            // constant trip: EXEC stays all-ones
                const int p  = pixbase + q;
                const int pc = (p < 49) ? p : 48;     // clamp; extra cols never stored
                const int py = pc / 7;
                const int sbase = 2 * pc + 14 * py;   // = 28*py + 2*px
                const float* ap = &S[sbase * SPITCH + lbase];
                v2f A = { ap[0], ap[SPITCH] };        // ds_load_2addr, offset1 = 65
                const float bval = (l15 == q) ? 0.25f : 0.0f;
                v2f B = { bval, bval };
                acc = __builtin_amdgcn_wmma_f32_16x16x4_f32(
                        false, A, false, B, (short)0, acc, false, false);
            }

            // D layout: VGPR r -> channel cbase + r + 8*half (M), pixel pixbase + l15 (N)
            const int pix = pixbase + l15;
            float* ob = out + ((size_t)box * 256 + ctglob + cbase + 8 * half) * 49 + pix;
            if (pix < 49) {                           // single per-lane guard, 8 stores
                #pragma unroll
                for (int r = 0; r < 8; ++r)
                    ob[(size_t)r * 49] = acc[r];
            }
        }
        __syncthreads();
    }
}

extern "C" void kernel_launch(void* const* d_in, const int* in_sizes, int n_in,
                              void* d_out, int out_size, void* d_ws, size_t ws_size,
                              hipStream_t stream) {
    const float* f0    = (const float*)d_in[0];
    const float* f1    = (const float*)d_in[1];
    const float* f2    = (const float*)d_in[2];
    const float* f3    = (const float*)d_in[3];
    const float* boxes = (const float*)d_in[4];
    const int*   bidx  = (const int*)d_in[5];
    float* out = (float*)d_out;

    const int nBoxes = in_sizes[4] / 4;
    roi_pool_fpn_wmma<<<dim3(nBoxes), dim3(256), 0, stream>>>(
        f0, f1, f2, f3, boxes, bidx, out, nBoxes);
}